// MACE_83348135346715
// MI455X (gfx1250) — compile-verified
//
#include <hip/hip_runtime.h>
#include <math.h>

#define NDIM 16
#define NRAD 8
#define RMAXF 5.0f
#define PI_F 3.14159265358979323846f

typedef __attribute__((ext_vector_type(2))) float v2f;
typedef __attribute__((ext_vector_type(8))) float v8f;

__device__ __forceinline__ void atomAddF(float* p, float v) { unsafeAtomicAdd(p, v); }

// ---------------- WMMA 16x16x16 f32 helpers (4 chained 16x16x4) ----------------
// A layout (ISA 7.12.2, 32-bit A 16x4): lane m = lane&15; lanes<16 hold K={0,1},
// lanes>=16 hold K={2,3} (per 4-wide K chunk). B mirrored: n = lane&15.
// D layout: VGPR j holds row j + 8*(lane>=16), col = lane&15.

__device__ __forceinline__ v8f wmma_mac16(const float* Arow, int astride,
                                          const float* B, bool btrans, v8f c) {
  int lane = threadIdx.x & 31;
  int koff = (lane >> 4) << 1;
  int n = lane & 15;
#pragma unroll
  for (int kc = 0; kc < 4; ++kc) {
    int k = kc * 4 + koff;
    v2f a, b;
    a.x = Arow[k * astride];
    a.y = Arow[(k + 1) * astride];
    if (btrans) { b.x = B[n * 16 + k]; b.y = B[n * 16 + k + 1]; }
    else        { b.x = B[k * 16 + n]; b.y = B[(k + 1) * 16 + n]; }
    c = __builtin_amdgcn_wmma_f32_16x16x4_f32(false, a, false, b, (short)0, c, false, false);
  }
  return c;
}

__device__ __forceinline__ v8f wmma_loadc(const float* in, int base, int Nn,
                                          int rstride, int cstride) {
  int lane = threadIdx.x & 31;
  int n = lane & 15;
  int r0 = (lane >> 4) << 3;
  v8f c;
  if (base + 16 <= Nn) {            // full tile: no clamping (wave-uniform)
#pragma unroll
    for (int j = 0; j < 8; ++j)
      c[j] = in[(size_t)(base + r0 + j) * rstride + n * cstride];
  } else {
#pragma unroll
    for (int j = 0; j < 8; ++j) {
      int row = base + r0 + j; if (row >= Nn) row = Nn - 1;
      c[j] = in[(size_t)row * rstride + n * cstride];
    }
  }
  return c;
}

__device__ __forceinline__ void wmma_store(float* out, int base, int Nn,
                                           int rstride, int cstride, v8f d) {
  int lane = threadIdx.x & 31;
  int n = lane & 15;
  int r0 = (lane >> 4) << 3;
  if (base + 16 <= Nn) {            // full tile: straight-line stores
#pragma unroll
    for (int j = 0; j < 8; ++j)
      out[(size_t)(base + r0 + j) * rstride + n * cstride] = d[j];
  } else {
#pragma unroll
    for (int j = 0; j < 8; ++j) {
      int row = base + r0 + j;
      if (row < Nn) out[(size_t)row * rstride + n * cstride] = d[j];
    }
  }
}

// ---------------- kernels ----------------

// Per-edge geometry: r, u, radial basis, SH l=1,2
__global__ void k_edge_geom(const float* __restrict__ pos, const int* __restrict__ ei, int E,
                            float* __restrict__ r_, float* __restrict__ u_,
                            float* __restrict__ rad_, float* __restrict__ sh1_,
                            float* __restrict__ sh2_) {
  int e = blockIdx.x * blockDim.x + threadIdx.x;
  if (e >= E) return;
  int row = ei[e], col = ei[E + e];
  float vx = pos[row * 3 + 0] - pos[col * 3 + 0];
  float vy = pos[row * 3 + 1] - pos[col * 3 + 1];
  float vz = pos[row * 3 + 2] - pos[col * 3 + 2];
  float r2 = vx * vx + vy * vy + vz * vz + 1e-12f;
  float r = sqrtf(r2);
  float ir = 1.0f / r;
  float x = vx * ir, y = vy * ir, z = vz * ir;
  r_[e] = r;
  u_[e * 3 + 0] = x; u_[e * 3 + 1] = y; u_[e * 3 + 2] = z;
  const float A = 0.632455532f; // sqrt(2/5)
  float4 rv0, rv1;
  rv0.x = A * sinf(1.0f * PI_F / RMAXF * r) * ir;
  rv0.y = A * sinf(2.0f * PI_F / RMAXF * r) * ir;
  rv0.z = A * sinf(3.0f * PI_F / RMAXF * r) * ir;
  rv0.w = A * sinf(4.0f * PI_F / RMAXF * r) * ir;
  rv1.x = A * sinf(5.0f * PI_F / RMAXF * r) * ir;
  rv1.y = A * sinf(6.0f * PI_F / RMAXF * r) * ir;
  rv1.z = A * sinf(7.0f * PI_F / RMAXF * r) * ir;
  rv1.w = A * sinf(8.0f * PI_F / RMAXF * r) * ir;
  float4* radv = (float4*)(rad_ + (size_t)e * NRAD);
  radv[0] = rv0; radv[1] = rv1;
  const float s3 = 1.7320508076f, s5 = 2.2360679775f, s15 = 3.8729833462f;
  sh1_[e * 3 + 0] = s3 * x; sh1_[e * 3 + 1] = s3 * y; sh1_[e * 3 + 2] = s3 * z;
  sh2_[(size_t)e * 5 + 0] = s15 * x * y;
  sh2_[(size_t)e * 5 + 1] = s15 * y * z;
  sh2_[(size_t)e * 5 + 2] = 0.5f * s5 * (3.0f * z * z - 1.0f);
  sh2_[(size_t)e * 5 + 3] = s15 * x * z;
  sh2_[(size_t)e * 5 + 4] = 0.5f * s15 * (x * x - y * y);
}

// h0 = emb[z] @ W_init  (gathered rows, WMMA)
__global__ void k_h0_init(const int* __restrict__ z, const float* __restrict__ emb,
                          const float* __restrict__ W_init, float* __restrict__ h0, int Nn) {
  int wave = blockIdx.x * (blockDim.x >> 5) + (threadIdx.x >> 5);
  int base = wave * 16;
  if (base >= Nn) return;
  int lane = threadIdx.x & 31;
  int rowc = base + (lane & 15); if (rowc >= Nn) rowc = Nn - 1;
  const float* Arow = emb + (size_t)z[rowc] * NDIM;
  v8f c = {};
  c = wmma_mac16(Arow, 1, W_init, false, c);
  wmma_store(h0, base, Nn, NDIM, 1, c);
}

// Per-edge messages + scatter (forward).  One thread per edge, loop over 16 channels.
template <bool HAS12>
__global__ void k_edge_msg(const int* __restrict__ ei, const float* __restrict__ rad_,
                           const float* __restrict__ sh1_, const float* __restrict__ sh2_,
                           const float* __restrict__ h0in, const float* __restrict__ h1in,
                           const float* __restrict__ h2in, const float* __restrict__ Wr,
                           const float* __restrict__ br, float* __restrict__ a0,
                           float* __restrict__ a1, float* __restrict__ a2, int E) {
  __shared__ float sWr[NRAD * 80];
  __shared__ float sbr[80];
  for (int i = threadIdx.x; i < NRAD * 80; i += blockDim.x) sWr[i] = Wr[i];
  for (int i = threadIdx.x; i < 80; i += blockDim.x) sbr[i] = br[i];
  __syncthreads();
  int e = blockIdx.x * blockDim.x + threadIdx.x;
  if (e >= E) return;
  int row = ei[e], col = ei[E + e];
  float rad[NRAD];
  {
    const float4* radv = (const float4*)(rad_ + (size_t)e * NRAD);
    float4 rv0 = radv[0], rv1 = radv[1];
    rad[0] = rv0.x; rad[1] = rv0.y; rad[2] = rv0.z; rad[3] = rv0.w;
    rad[4] = rv1.x; rad[5] = rv1.y; rad[6] = rv1.z; rad[7] = rv1.w;
  }
  float s1[3], s2[5];
#pragma unroll
  for (int m = 0; m < 3; ++m) s1[m] = sh1_[e * 3 + m];
#pragma unroll
  for (int m = 0; m < 5; ++m) s2[m] = sh2_[(size_t)e * 5 + m];
  float h0c[NDIM];
  {
    const float4* hv = (const float4*)(h0in + (size_t)col * NDIM);
#pragma unroll
    for (int q = 0; q < 4; ++q) {
      float4 v = hv[q];
      h0c[q * 4 + 0] = v.x; h0c[q * 4 + 1] = v.y; h0c[q * 4 + 2] = v.z; h0c[q * 4 + 3] = v.w;
    }
  }
  for (int c = 0; c < NDIM; ++c) {
    float S1 = 0.f, S2 = 0.f;
    if (HAS12) {
#pragma unroll
      for (int m = 0; m < 3; ++m) S1 += h1in[(size_t)col * 48 + c * 3 + m] * s1[m];
#pragma unroll
      for (int m = 0; m < 5; ++m) S2 += h2in[(size_t)col * 80 + c * 5 + m] * s2[m];
    }
    float wp[5];
#pragma unroll
    for (int p = 0; p < 5; ++p) {
      float acc = sbr[p * 16 + c];
#pragma unroll
      for (int n = 0; n < NRAD; ++n) acc += rad[n] * sWr[n * 80 + p * 16 + c];
      wp[p] = acc;
    }
    atomAddF(&a0[(size_t)row * NDIM + c], wp[0] * h0c[c] + wp[3] * S1 + wp[4] * S2);
    float t1 = wp[1] * h0c[c], t2 = wp[2] * h0c[c];
#pragma unroll
    for (int m = 0; m < 3; ++m) atomAddF(&a1[(size_t)row * 48 + c * 3 + m], t1 * s1[m]);
#pragma unroll
    for (int m = 0; m < 5; ++m) atomAddF(&a2[(size_t)row * 80 + c * 5 + m], t2 * s2[m]);
  }
}

// h0' = h0 + a0@W0 ; h1' = a1 (x) W1 ; h2' = a2 (x) W2   (WMMA, per-m GEMMs)
template <bool FULL>
__global__ void k_node_update(const float* __restrict__ h0_in, const float* __restrict__ a0,
                              const float* __restrict__ a1, const float* __restrict__ a2,
                              const float* __restrict__ Wout, float* __restrict__ h0_out,
                              float* __restrict__ h1_out, float* __restrict__ h2_out, int Nn) {
  int wave = blockIdx.x * (blockDim.x >> 5) + (threadIdx.x >> 5);
  int base = wave * 16;
  if (base >= Nn) return;
  int lane = threadIdx.x & 31;
  int rowc = base + (lane & 15); if (rowc >= Nn) rowc = Nn - 1;
  v8f c = wmma_loadc(h0_in, base, Nn, NDIM, 1);
  c = wmma_mac16(a0 + (size_t)rowc * NDIM, 1, Wout, false, c);
  wmma_store(h0_out, base, Nn, NDIM, 1, c);
  if (FULL) {
#pragma unroll
    for (int mm = 0; mm < 3; ++mm) {
      v8f c1 = {};
      c1 = wmma_mac16(a1 + (size_t)rowc * 48 + mm, 3, Wout + 256, false, c1);
      wmma_store(h1_out + mm, base, Nn, 48, 3, c1);
    }
#pragma unroll
    for (int mm = 0; mm < 5; ++mm) {
      v8f c2 = {};
      c2 = wmma_mac16(a2 + (size_t)rowc * 80 + mm, 5, Wout + 512, false, c2);
      wmma_store(h2_out + mm, base, Nn, 80, 5, c2);
    }
  }
}

// t = (h0 @ W_read) @ W1 + b1   (two chained WMMAs through LDS)
__global__ void k_readout_fwd(const float* __restrict__ h0, const float* __restrict__ W_read,
                              const float* __restrict__ W1, const float* __restrict__ b1,
                              float* __restrict__ t_out, int Nn) {
  __shared__ float lds[8][16][17];
  int wv = threadIdx.x >> 5;
  int base = (blockIdx.x * 8 + wv) * 16;
  bool active = base < Nn;
  int lane = threadIdx.x & 31;
  int m = lane & 15;
  v8f c = {};
  if (active) {
    int rowc = base + m; if (rowc >= Nn) rowc = Nn - 1;
    c = wmma_mac16(h0 + (size_t)rowc * NDIM, 1, W_read, false, c);
  }
  int r0 = (lane >> 4) << 3;
#pragma unroll
  for (int j = 0; j < 8; ++j) lds[wv][r0 + j][m] = c[j];
  __syncthreads();
  if (active) {
    float bv = b1[m];
    v8f c2;
#pragma unroll
    for (int j = 0; j < 8; ++j) c2[j] = bv;
    c2 = wmma_mac16(&lds[wv][m][0], 1, W1, false, c2);
    wmma_store(t_out, base, Nn, NDIM, 1, c2);
  }
}

// site = silu(t) @ W2 + b2 ; energy = sum(site)
__global__ void k_site(const float* __restrict__ t, const float* __restrict__ W2,
                       const float* __restrict__ b2, float* __restrict__ energy, int Nn) {
  __shared__ float red[256];
  int ni = blockIdx.x * blockDim.x + threadIdx.x;
  float acc = 0.f;
  if (ni < Nn) {
#pragma unroll
    for (int d = 0; d < NDIM; ++d) {
      float tv = t[(size_t)ni * NDIM + d];
      float s = tv / (1.0f + expf(-tv));
      acc += s * W2[d];
    }
    acc += b2[0];
  }
  red[threadIdx.x] = acc;
  __syncthreads();
  for (int s = 128; s > 0; s >>= 1) {
    if (threadIdx.x < s) red[threadIdx.x] += red[threadIdx.x + s];
    __syncthreads();
  }
  if (threadIdx.x == 0) atomAddF(energy, red[0]);
}

// g0 = ((W2 * silu'(t)) @ W1^T) @ W_read^T
__global__ void k_readout_bwd(const float* __restrict__ t, const float* __restrict__ W_read,
                              const float* __restrict__ W1, const float* __restrict__ W2,
                              float* __restrict__ g0, int Nn) {
  __shared__ float lds[8][16][17];
  int wv = threadIdx.x >> 5;
  int base = (blockIdx.x * 8 + wv) * 16;
  bool active = base < Nn;
  int lane = threadIdx.x & 31;
  int m = lane & 15;
  int koff = (lane >> 4) << 1;
  v8f c = {};
  if (active) {
    int rowc = base + m; if (rowc >= Nn) rowc = Nn - 1;
    const float* trow = t + (size_t)rowc * NDIM;
#pragma unroll
    for (int kc = 0; kc < 4; ++kc) {
      int k = kc * 4 + koff;
      float t0 = trow[k], t1 = trow[k + 1];
      float sg0 = 1.0f / (1.0f + expf(-t0));
      float sg1 = 1.0f / (1.0f + expf(-t1));
      v2f a, b;
      a.x = W2[k]     * sg0 * (1.0f + t0 * (1.0f - sg0));
      a.y = W2[k + 1] * sg1 * (1.0f + t1 * (1.0f - sg1));
      b.x = W1[m * 16 + k]; b.y = W1[m * 16 + k + 1];  // W1^T
      c = __builtin_amdgcn_wmma_f32_16x16x4_f32(false, a, false, b, (short)0, c, false, false);
    }
  }
  int r0 = (lane >> 4) << 3;
#pragma unroll
  for (int j = 0; j < 8; ++j) lds[wv][r0 + j][m] = c[j];
  __syncthreads();
  if (active) {
    v8f c2 = {};
    c2 = wmma_mac16(&lds[wv][m][0], 1, W_read, true, c2);
    wmma_store(g0, base, Nn, NDIM, 1, c2);
  }
}

// da0 = g0 @ W0^T ; da1 = g1 (x) W1^T ; da2 = g2 (x) W2^T
template <bool FULL>
__global__ void k_node_grad(const float* __restrict__ g0, const float* __restrict__ g1,
                            const float* __restrict__ g2, const float* __restrict__ Wout,
                            float* __restrict__ da0, float* __restrict__ da1,
                            float* __restrict__ da2, int Nn) {
  int wave = blockIdx.x * (blockDim.x >> 5) + (threadIdx.x >> 5);
  int base = wave * 16;
  if (base >= Nn) return;
  int lane = threadIdx.x & 31;
  int rowc = base + (lane & 15); if (rowc >= Nn) rowc = Nn - 1;
  v8f c = {};
  c = wmma_mac16(g0 + (size_t)rowc * NDIM, 1, Wout, true, c);
  wmma_store(da0, base, Nn, NDIM, 1, c);
  if (FULL) {
#pragma unroll
    for (int mm = 0; mm < 3; ++mm) {
      v8f c1 = {};
      c1 = wmma_mac16(g1 + (size_t)rowc * 48 + mm, 3, Wout + 256, true, c1);
      wmma_store(da1 + mm, base, Nn, 48, 3, c1);
    }
#pragma unroll
    for (int mm = 0; mm < 5; ++mm) {
      v8f c2 = {};
      c2 = wmma_mac16(g2 + (size_t)rowc * 80 + mm, 5, Wout + 512, true, c2);
      wmma_store(da2 + mm, base, Nn, 80, 5, c2);
    }
  }
}

// Edge backward: dh scatter into g*, dsh/dradial -> forces
template <bool HAS12, bool HASDM12, bool SCATG>
__global__ void k_edge_bwd(const int* __restrict__ ei, const float* __restrict__ r_,
                           const float* __restrict__ u_, const float* __restrict__ rad_,
                           const float* __restrict__ sh1_, const float* __restrict__ sh2_,
                           const float* __restrict__ h0in, const float* __restrict__ h1in,
                           const float* __restrict__ h2in, const float* __restrict__ da0,
                           const float* __restrict__ da1, const float* __restrict__ da2,
                           const float* __restrict__ Wr, const float* __restrict__ br,
                           float* __restrict__ g0, float* __restrict__ g1, float* __restrict__ g2,
                           float* __restrict__ force, int E) {
  __shared__ float sWr[NRAD * 80];
  __shared__ float sbr[80];
  for (int i = threadIdx.x; i < NRAD * 80; i += blockDim.x) sWr[i] = Wr[i];
  for (int i = threadIdx.x; i < 80; i += blockDim.x) sbr[i] = br[i];
  __syncthreads();
  int e = blockIdx.x * blockDim.x + threadIdx.x;
  if (e >= E) return;
  int row = ei[e], col = ei[E + e];
  float rad[NRAD];
  {
    const float4* radv = (const float4*)(rad_ + (size_t)e * NRAD);
    float4 rv0 = radv[0], rv1 = radv[1];
    rad[0] = rv0.x; rad[1] = rv0.y; rad[2] = rv0.z; rad[3] = rv0.w;
    rad[4] = rv1.x; rad[5] = rv1.y; rad[6] = rv1.z; rad[7] = rv1.w;
  }
  float s1[3], s2[5];
#pragma unroll
  for (int m = 0; m < 3; ++m) s1[m] = sh1_[e * 3 + m];
#pragma unroll
  for (int m = 0; m < 5; ++m) s2[m] = sh2_[(size_t)e * 5 + m];
  float dm0v[NDIM], h0cv[NDIM];
  {
    const float4* dv = (const float4*)(da0 + (size_t)row * NDIM);
    const float4* hv = (const float4*)(h0in + (size_t)col * NDIM);
#pragma unroll
    for (int q = 0; q < 4; ++q) {
      float4 a = dv[q], b = hv[q];
      dm0v[q * 4 + 0] = a.x; dm0v[q * 4 + 1] = a.y; dm0v[q * 4 + 2] = a.z; dm0v[q * 4 + 3] = a.w;
      h0cv[q * 4 + 0] = b.x; h0cv[q * 4 + 1] = b.y; h0cv[q * 4 + 2] = b.z; h0cv[q * 4 + 3] = b.w;
    }
  }
  float drad[NRAD] = {0.f}, dsh1[3] = {0.f}, dsh2[5] = {0.f};
  for (int c = 0; c < NDIM; ++c) {
    float dm0 = dm0v[c];
    float h0c = h0cv[c];
    float dm1[3] = {0.f, 0.f, 0.f}, dm2[5] = {0.f, 0.f, 0.f, 0.f, 0.f};
    if (HASDM12) {
#pragma unroll
      for (int m = 0; m < 3; ++m) dm1[m] = da1[(size_t)row * 48 + c * 3 + m];
#pragma unroll
      for (int m = 0; m < 5; ++m) dm2[m] = da2[(size_t)row * 80 + c * 5 + m];
    }
    float h1c[3] = {0.f, 0.f, 0.f}, h2c[5] = {0.f, 0.f, 0.f, 0.f, 0.f};
    if (HAS12) {
#pragma unroll
      for (int m = 0; m < 3; ++m) h1c[m] = h1in[(size_t)col * 48 + c * 3 + m];
#pragma unroll
      for (int m = 0; m < 5; ++m) h2c[m] = h2in[(size_t)col * 80 + c * 5 + m];
    }
    float wp[5];
#pragma unroll
    for (int p = 0; p < 5; ++p) {
      float acc = sbr[p * 16 + c];
#pragma unroll
      for (int n = 0; n < NRAD; ++n) acc += rad[n] * sWr[n * 80 + p * 16 + c];
      wp[p] = acc;
    }
    float S1 = 0.f, S2 = 0.f, T1 = 0.f, T2 = 0.f;
#pragma unroll
    for (int m = 0; m < 3; ++m) { S1 += h1c[m] * s1[m]; T1 += dm1[m] * s1[m]; }
#pragma unroll
    for (int m = 0; m < 5; ++m) { S2 += h2c[m] * s2[m]; T2 += dm2[m] * s2[m]; }
    float dw0 = dm0 * h0c, dw1 = h0c * T1, dw2 = h0c * T2, dw3 = dm0 * S1, dw4 = dm0 * S2;
#pragma unroll
    for (int n = 0; n < NRAD; ++n)
      drad[n] += dw0 * sWr[n * 80 + c] + dw1 * sWr[n * 80 + 16 + c] + dw2 * sWr[n * 80 + 32 + c] +
                 dw3 * sWr[n * 80 + 48 + c] + dw4 * sWr[n * 80 + 64 + c];
    if (SCATG) {
      atomAddF(&g0[(size_t)col * NDIM + c], dm0 * wp[0] + wp[1] * T1 + wp[2] * T2);
      if (HAS12) {
#pragma unroll
        for (int m = 0; m < 3; ++m) atomAddF(&g1[(size_t)col * 48 + c * 3 + m], dm0 * wp[3] * s1[m]);
#pragma unroll
        for (int m = 0; m < 5; ++m) atomAddF(&g2[(size_t)col * 80 + c * 5 + m], dm0 * wp[4] * s2[m]);
      }
    }
#pragma unroll
    for (int m = 0; m < 3; ++m) dsh1[m] += dm0 * wp[3] * h1c[m] + dm1[m] * wp[1] * h0c;
#pragma unroll
    for (int m = 0; m < 5; ++m) dsh2[m] += dm0 * wp[4] * h2c[m] + dm2[m] * wp[2] * h0c;
  }
  // geometry chain: radial(r), sh(u) -> v -> pos
  float r = r_[e];
  float ir = 1.0f / r;
  const float A = 0.632455532f;
  float dr = 0.f;
#pragma unroll
  for (int nn = 1; nn <= NRAD; ++nn) {
    float kn = (float)nn * PI_F / RMAXF;
    dr += drad[nn - 1] * A * (kn * cosf(kn * r) * ir - sinf(kn * r) * ir * ir);
  }
  float x = u_[e * 3 + 0], y = u_[e * 3 + 1], z = u_[e * 3 + 2];
  const float s3 = 1.7320508076f, s5 = 2.2360679775f, s15 = 3.8729833462f;
  float dux = s3 * dsh1[0] + s15 * (dsh2[0] * y + dsh2[3] * z) + s15 * x * dsh2[4];
  float duy = s3 * dsh1[1] + s15 * (dsh2[0] * x + dsh2[1] * z) - s15 * y * dsh2[4];
  float duz = s3 * dsh1[2] + s15 * (dsh2[1] * y + dsh2[3] * x) + 3.0f * s5 * z * dsh2[2];
  float udd = x * dux + y * duy + z * duz;
  float dvx = (dux - x * udd) * ir + dr * x;
  float dvy = (duy - y * udd) * ir + dr * y;
  float dvz = (duz - z * udd) * ir + dr * z;
  // forces = -dE/dpos ; pos[row] gets +dv, pos[col] gets -dv
  atomAddF(&force[row * 3 + 0], -dvx);
  atomAddF(&force[row * 3 + 1], -dvy);
  atomAddF(&force[row * 3 + 2], -dvz);
  atomAddF(&force[col * 3 + 0], dvx);
  atomAddF(&force[col * 3 + 1], dvy);
  atomAddF(&force[col * 3 + 2], dvz);
}

// ---------------- host launch ----------------
extern "C" void kernel_launch(void* const* d_in, const int* in_sizes, int n_in,
                              void* d_out, int out_size, void* d_ws, size_t ws_size,
                              hipStream_t stream) {
  const float* pos    = (const float*)d_in[0];
  const int*   z      = (const int*)d_in[1];
  const int*   ei     = (const int*)d_in[2];
  const float* emb    = (const float*)d_in[3];
  const float* W_init = (const float*)d_in[4];
  const float* Wr     = (const float*)d_in[5];
  const float* br     = (const float*)d_in[6];
  const float* W_out  = (const float*)d_in[7];
  const float* W_read = (const float*)d_in[8];
  const float* W1     = (const float*)d_in[9];
  const float* b1     = (const float*)d_in[10];
  const float* W2     = (const float*)d_in[11];
  const float* b2     = (const float*)d_in[12];
  int Nn = in_sizes[0] / 3;
  int E  = in_sizes[2] / 2;

  float* p = (float*)d_ws;
  float* r_    = p; p += E;
  float* u_    = p; p += (size_t)3 * E;
  float* rad_  = p; p += (size_t)NRAD * E;
  float* sh1_  = p; p += (size_t)3 * E;
  float* sh2_  = p; p += (size_t)5 * E;
  float* h0_s0 = p; p += (size_t)16 * Nn;
  float* h0_s1 = p; p += (size_t)16 * Nn;
  float* h0_s2 = p; p += (size_t)16 * Nn;
  float* h1_s1 = p; p += (size_t)48 * Nn;
  float* h2_s1 = p; p += (size_t)80 * Nn;
  float* a0    = p; p += (size_t)16 * Nn;   // a0,a1,a2 contiguous (single memset)
  float* a1    = p; p += (size_t)48 * Nn;
  float* a2    = p; p += (size_t)80 * Nn;
  float* tbuf  = p; p += (size_t)16 * Nn;
  float* g0    = p; p += (size_t)16 * Nn;
  float* g1    = p; p += (size_t)48 * Nn;   // g1,g2 contiguous (single memset)
  float* g2    = p; p += (size_t)80 * Nn;

  float* energy = (float*)d_out;
  float* forces = (float*)d_out + 1;

  int ethreads = 256;
  int eblocks  = (E + ethreads - 1) / ethreads;
  int tiles    = (Nn + 15) / 16;
  int tblocks  = (tiles + 7) / 8;       // 8 waves (tiles) per 256-thread block
  int nblocks  = (Nn + 255) / 256;

  hipMemsetAsync(d_out, 0, sizeof(float) * (size_t)(1 + 3 * Nn), stream);

  // forward
  k_edge_geom<<<eblocks, ethreads, 0, stream>>>(pos, ei, E, r_, u_, rad_, sh1_, sh2_);
  k_h0_init<<<tblocks, 256, 0, stream>>>(z, emb, W_init, h0_s0, Nn);

  hipMemsetAsync(a0, 0, sizeof(float) * (size_t)144 * Nn, stream);
  k_edge_msg<false><<<eblocks, ethreads, 0, stream>>>(ei, rad_, sh1_, sh2_, h0_s0, nullptr,
                                                      nullptr, Wr, br, a0, a1, a2, E);
  k_node_update<true><<<tblocks, 256, 0, stream>>>(h0_s0, a0, a1, a2, W_out,
                                                   h0_s1, h1_s1, h2_s1, Nn);

  hipMemsetAsync(a0, 0, sizeof(float) * (size_t)144 * Nn, stream);
  k_edge_msg<true><<<eblocks, ethreads, 0, stream>>>(ei, rad_, sh1_, sh2_, h0_s1, h1_s1, h2_s1,
                                                     Wr + 640, br + 80, a0, a1, a2, E);
  k_node_update<false><<<tblocks, 256, 0, stream>>>(h0_s1, a0, a1, a2, W_out + 768, h0_s2,
                                                    nullptr, nullptr, Nn);

  k_readout_fwd<<<tblocks, 256, 0, stream>>>(h0_s2, W_read, W1, b1, tbuf, Nn);
  k_site<<<nblocks, 256, 0, stream>>>(tbuf, W2, b2, energy, Nn);

  // backward
  k_readout_bwd<<<tblocks, 256, 0, stream>>>(tbuf, W_read, W1, W2, g0, Nn);
  hipMemsetAsync(g1, 0, sizeof(float) * (size_t)128 * Nn, stream);

  // block 1 backward (g1,g2 zero at block-1 output -> da1=da2=0)
  k_node_grad<false><<<tblocks, 256, 0, stream>>>(g0, g1, g2, W_out + 768, a0, a1, a2, Nn);
  k_edge_bwd<true, false, true><<<eblocks, ethreads, 0, stream>>>(
      ei, r_, u_, rad_, sh1_, sh2_, h0_s1, h1_s1, h2_s1, a0, a1, a2,
      Wr + 640, br + 80, g0, g1, g2, forces, E);
  // block 0 backward (h1/h2 inputs are zero; upstream h grads don't reach pos)
  k_node_grad<true><<<tblocks, 256, 0, stream>>>(g0, g1, g2, W_out, a0, a1, a2, Nn);
  k_edge_bwd<false, true, false><<<eblocks, ethreads, 0, stream>>>(
      ei, r_, u_, rad_, sh1_, sh2_, h0_s0, nullptr, nullptr, a0, a1, a2,
      Wr, br, g0, g1, g2, forces, E);
}